// CVLoss_80376017977765
// MI455X (gfx1250) — compile-verified
//
#include <hip/hip_runtime.h>

// Problem constants (match reference: B=8, T=2000, N=2048)
constexpr int Bb = 8;
constexpr int Tt = 2000;
constexpr int Nn = 2048;
constexpr float PENALTY = 10.0f;

constexpr int TSPLIT = 4;           // time segments per column (parallelism)
constexpr int TSEG   = Tt / TSPLIT; // 500
constexpr int TCHUNK = 20;          // timesteps staged per LDS chunk
constexpr int NCHUNK = TSEG / TCHUNK; // 25
constexpr int NTILE  = 256;         // neurons per block
constexpr int THREADS = 256;

typedef float v8f __attribute__((ext_vector_type(8)));
typedef float v2f __attribute__((ext_vector_type(2)));

// ---------------------------------------------------------------------------
// Kernel 1: streaming scan. Each block: one (b, n-tile of 256, T-segment).
// Chunks of [TCHUNK x 256] f32 are DMA'd into LDS with async-to-LDS loads
// (double buffered, ASYNCcnt), then each thread scans its neuron column.
// Per-segment monoid state {count, first, last, s2} written to workspace.
// ---------------------------------------------------------------------------
__global__ __launch_bounds__(THREADS) void cv_scan_kernel(
    const float* __restrict__ spikes,
    float* __restrict__ segCnt, float* __restrict__ segFirst,
    float* __restrict__ segLast, float* __restrict__ segS2)
{
    __shared__ float tile[2][TCHUNK * NTILE];   // 2 x 20KB = 40KB

    const int tid   = threadIdx.x;
    const int nBase = blockIdx.x * NTILE;
    const int b     = blockIdx.y;
    const int seg   = blockIdx.z;
    const int t0    = seg * TSEG;

    const float* base = spikes + ((size_t)b * Tt + (size_t)t0) * Nn + nBase;

    // Issue one chunk (TCHUNK rows x 1024B) as async global->LDS b128 copies.
    auto issue = [&](int chunk, int buf) {
        unsigned ldsBase = (unsigned)(size_t)(&tile[buf][0]);
        const float* gch = base + (size_t)chunk * TCHUNK * Nn;
        #pragma unroll
        for (int i = 0; i < (TCHUNK * 64) / THREADS; ++i) {   // 5 x b128 / thread
            int c     = tid + THREADS * i;   // 16B-chunk id within tile
            int row   = c >> 6;              // 64 chunks per 1024B row
            int col16 = c & 63;
            unsigned lds = ldsBase + (unsigned)c * 16u;
            unsigned long long ga =
                (unsigned long long)(size_t)(gch + (size_t)row * Nn) +
                (unsigned long long)(col16 * 16);
            asm volatile("global_load_async_to_lds_b128 %0, %1, off"
                         :: "v"(lds), "v"(ga) : "memory");
        }
    };

    float last = -1.0f, first = -1.0f, s2 = 0.0f, cnt = 0.0f;

    issue(0, 0);
    asm volatile("s_wait_asynccnt 0x0" ::: "memory");
    __syncthreads();

    for (int k = 0; k < NCHUNK; ++k) {
        const int buf = k & 1;
        if (k + 1 < NCHUNK) issue(k + 1, buf ^ 1);   // overlap DMA with scan

        const float* sm = &tile[buf][0];
        const float tf = (float)(t0 + k * TCHUNK);
        #pragma unroll
        for (int t = 0; t < TCHUNK; ++t) {
            float x    = sm[t * NTILE + tid];        // conflict-free ds_load
            float tcur = tf + (float)t;
            if (x > 0.0f) {
                float d = tcur - last;
                s2   += (last >= 0.0f) ? d * d : 0.0f;
                first = (first < 0.0f) ? tcur : first;
                last  = tcur;
                cnt  += 1.0f;
            }
        }

        if (k + 1 < NCHUNK) asm volatile("s_wait_asynccnt 0x0" ::: "memory");
        __syncthreads();   // chunk k+1 visible to all waves; buf free for reuse
    }

    const size_t idx = ((size_t)seg * Bb + b) * Nn + nBase + tid;
    segCnt[idx]   = cnt;
    segFirst[idx] = first;
    segLast[idx]  = last;
    segS2[idx]    = s2;
}

// ---------------------------------------------------------------------------
// Wave-level sum of 64 shared floats using the matrix unit:
// ones(16x4) x B(4x16) + C  ->  every D row holds the 16 column sums.
// Caller: exactly one full wave (lanes 0..31), EXEC all ones.
// ---------------------------------------------------------------------------
__device__ inline float wave_wmma_sum64(const float* vals, int lane) {
    const int nlo = lane & 15;
    const int hi  = (lane >= 16) ? 32 : 0;
    v2f a; a.x = 1.0f; a.y = 1.0f;              // all-ones A (16x4)
    v2f bm;                                     // B(4x16): V0 = K0/K2, V1 = K1/K3
    bm.x = vals[hi + nlo];
    bm.y = vals[hi + 16 + nlo];
    v8f c8 = {};
    c8 = __builtin_amdgcn_wmma_f32_16x16x4_f32(false, a, false, bm,
                                               (short)0, c8, false, false);
    float v = c8[0];                            // lane -> colsum(lane & 15)
    v += __shfl_xor(v, 1, 32);
    v += __shfl_xor(v, 2, 32);
    v += __shfl_xor(v, 4, 32);
    v += __shfl_xor(v, 8, 32);
    return v;                                   // total in every lane
}

// ---------------------------------------------------------------------------
// Kernel 2: merge TSPLIT segment-states per (b,n), compute CV + squared error,
// then reduce the block's 256 errors to one partial (WMMA reduction by wave 0).
// ---------------------------------------------------------------------------
__global__ __launch_bounds__(256) void cv_merge_kernel(
    const float* __restrict__ segCnt, const float* __restrict__ segFirst,
    const float* __restrict__ segLast, const float* __restrict__ segS2,
    const float* __restrict__ target, float* __restrict__ partial)
{
    const int tid = threadIdx.x;
    const int gi  = blockIdx.x * 256 + tid;     // flat (b,n)
    const int n   = gi & (Nn - 1);

    // Sequential (deterministic) monoid merge over the 4 time segments.
    float cnt = 0.0f, first = -1.0f, last = -1.0f, s2 = 0.0f;
    #pragma unroll
    for (int s = 0; s < TSPLIT; ++s) {
        const size_t idx = (size_t)s * (Bb * Nn) + gi;
        float c = segCnt[idx];
        if (c > 0.0f) {
            float f = segFirst[idx];
            if (cnt > 0.0f) {
                float d = f - last;             // boundary ISI
                s2 += segS2[idx] + d * d;
            } else {
                first = f;
                s2 += segS2[idx];
            }
            last = segLast[idx];
            cnt += c;
        }
    }

    const float k    = fmaxf(cnt - 1.0f, 0.0f);       // #ISIs
    const float s1   = (cnt > 0.0f) ? (last - first) : 0.0f;  // telescoped sum
    const float mean = s1 / fmaxf(k, 1.0f);
    const float var  = (s2 - k * mean * mean) / fmaxf(k - 1.0f, 1.0f);
    const float stdv = sqrtf(fmaxf(var, 0.0f));
    const float cv   = stdv / ((mean > 0.0f) ? mean : 1.0f);
    const float cvs  = ((cnt >= 3.0f) && (mean > 0.0f)) ? cv : PENALTY;
    const float diff = cvs - target[n];
    const float err  = diff * diff;

    __shared__ float red[256];
    red[tid] = err;
    __syncthreads();

    if (tid < 32) {                              // wave 0, full EXEC
        float acc = 0.0f;
        #pragma unroll
        for (int ch = 0; ch < 4; ++ch)
            acc += wave_wmma_sum64(&red[ch * 64], tid);
        if (tid == 0) partial[blockIdx.x] = acc;
    }
}

// ---------------------------------------------------------------------------
// Kernel 3: one wave reduces the 64 block partials (single WMMA) and scales.
// ---------------------------------------------------------------------------
__global__ __launch_bounds__(32) void cv_final_kernel(
    const float* __restrict__ partial, float* __restrict__ out)
{
    const int lane = threadIdx.x;
    float total = wave_wmma_sum64(partial, lane);
    if (lane == 0) out[0] = total * (1.0f / (float)(Bb * Nn));
}

// ---------------------------------------------------------------------------
extern "C" void kernel_launch(void* const* d_in, const int* in_sizes, int n_in,
                              void* d_out, int out_size, void* d_ws, size_t ws_size,
                              hipStream_t stream) {
    const float* spikes = (const float*)d_in[0];   // (B,T,N) f32
    const float* target = (const float*)d_in[1];   // (N,) f32
    float* out = (float*)d_out;

    float* ws       = (float*)d_ws;
    const size_t SEG = (size_t)TSPLIT * Bb * Nn;   // 65536 floats per array
    float* segCnt   = ws;
    float* segFirst = ws + SEG;
    float* segLast  = ws + 2 * SEG;
    float* segS2    = ws + 3 * SEG;
    float* partial  = ws + 4 * SEG;                // 64 floats

    dim3 g1(Nn / NTILE, Bb, TSPLIT);               // (8, 8, 4) = 256 blocks
    cv_scan_kernel<<<g1, THREADS, 0, stream>>>(spikes, segCnt, segFirst,
                                               segLast, segS2);

    const int nCols = Bb * Nn;                     // 16384
    cv_merge_kernel<<<nCols / 256, 256, 0, stream>>>(segCnt, segFirst, segLast,
                                                     segS2, target, partial);

    cv_final_kernel<<<1, 32, 0, stream>>>(partial, out);
}